// PZMsFeatureExtractor_17832704213330
// MI455X (gfx1250) — compile-verified
//
#include <hip/hip_runtime.h>
#include <math.h>

// Problem constants (match reference: x = (32, 3, 512, 512) fp32)
#define BIMG 32
#define HH   512
#define WW   512
#define NPX  (HH * WW)        // 262144 pixels per image
#define NBLK 64               // blocks per image for pixel passes
#define PXB  (NPX / NBLK)     // 4096 pixels per block
#define TPB  256

typedef __attribute__((ext_vector_type(2))) float v2f;
typedef __attribute__((ext_vector_type(8))) float v8f;

// ---------------------------------------------------------------------------
// Deterministic block-wide sum: wave32 shfl_xor butterfly + LDS across waves.
// ---------------------------------------------------------------------------
__device__ __forceinline__ float blockReduce(float v, float* smem) {
    const int lane = threadIdx.x & 31;
    const int wid  = threadIdx.x >> 5;
    const int nw   = blockDim.x >> 5;
#pragma unroll
    for (int m = 16; m >= 1; m >>= 1) v += __shfl_xor(v, m, 32);
    if (lane == 0) smem[wid] = v;
    __syncthreads();
    float r = 0.f;
    if (wid == 0) {
        r = (lane < nw) ? smem[lane] : 0.f;
#pragma unroll
        for (int m = 16; m >= 1; m >>= 1) r += __shfl_xor(r, m, 32);
    }
    __syncthreads();
    return r;   // full sum valid in wave 0 (lane 0 used)
}

__device__ __forceinline__ float gray4(float r, float g, float b) {
    return 0.299f * r + 0.587f * g + 0.114f * b;
}

// ---------------------------------------------------------------------------
// Pass A: gray conversion + per-block (sum, sumsq) partials.
// grid (NBLK, BIMG), block TPB. float4 coalesced loads of the 3 planes.
// ---------------------------------------------------------------------------
__global__ void k_gray_stats(const float* __restrict__ x, float* __restrict__ partA) {
    __shared__ float smem[TPB / 32];
    const int img = blockIdx.y;
    const float* Rp = x + (size_t)img * 3 * NPX;
    const float* Gp = Rp + NPX;
    const float* Bp = Rp + 2 * NPX;
    const int base = blockIdx.x * PXB;

    float s = 0.f, s2 = 0.f;
#pragma unroll
    for (int i = 0; i < 4; ++i) {
        const int p = base + i * 1024 + threadIdx.x * 4;
        float4 r4 = *(const float4*)(Rp + p);
        float4 g4 = *(const float4*)(Gp + p);
        float4 b4 = *(const float4*)(Bp + p);
        float g0 = gray4(r4.x, g4.x, b4.x);
        float g1 = gray4(r4.y, g4.y, b4.y);
        float g2 = gray4(r4.z, g4.z, b4.z);
        float g3 = gray4(r4.w, g4.w, b4.w);
        s  += (g0 + g1) + (g2 + g3);
        s2 += (g0 * g0 + g1 * g1) + (g2 * g2 + g3 * g3);
    }
    s  = blockReduce(s,  smem);
    s2 = blockReduce(s2, smem);
    if (threadIdx.x == 0) {
        float* o = partA + ((size_t)img * NBLK + blockIdx.x) * 2;
        o[0] = s; o[1] = s2;
    }
}

// ---------------------------------------------------------------------------
// Stats: mean, 1/(std+1e-8) per image. grid BIMG, block 64 (one thread/block-partial).
// ---------------------------------------------------------------------------
__global__ void k_stats(const float* __restrict__ partA, float* __restrict__ stats) {
    __shared__ float smem[2];
    const int img = blockIdx.x;
    const float* p = partA + (size_t)img * NBLK * 2;
    float s  = p[threadIdx.x * 2 + 0];
    float s2 = p[threadIdx.x * 2 + 1];
    s  = blockReduce(s,  smem);
    s2 = blockReduce(s2, smem);
    if (threadIdx.x == 0) {
        float mean = s / (float)NPX;
        float var  = s2 / (float)NPX - mean * mean;
        float sd   = sqrtf(fmaxf(var, 0.f));
        stats[img * 2 + 0] = mean;
        stats[img * 2 + 1] = 1.f / (sd + 1e-8f);
    }
}

// ---------------------------------------------------------------------------
// Pass B: raw moments of normalized image: 9 partials per block.
// [ S, Sx, Sy, Sxx, Sxy, Syy, Sa, Sax, Say ]
// ---------------------------------------------------------------------------
__global__ void k_moments(const float* __restrict__ x, const float* __restrict__ stats,
                          float* __restrict__ partB) {
    __shared__ float smem[TPB / 32];
    const int img = blockIdx.y;
    const float mean = stats[img * 2 + 0];
    const float istd = stats[img * 2 + 1];
    const float* Rp = x + (size_t)img * 3 * NPX;
    const float* Gp = Rp + NPX;
    const float* Bp = Rp + 2 * NPX;
    const int base = blockIdx.x * PXB;

    float a0 = 0.f, a1 = 0.f, a2 = 0.f, a3 = 0.f, a4 = 0.f,
          a5 = 0.f, a6 = 0.f, a7 = 0.f, a8 = 0.f;
#pragma unroll
    for (int i = 0; i < 4; ++i) {
        const int p = base + i * 1024 + threadIdx.x * 4;
        const float fy = (float)(p >> 9);
        const float fx0 = (float)(p & 511);
        float4 r4 = *(const float4*)(Rp + p);
        float4 g4 = *(const float4*)(Gp + p);
        float4 b4 = *(const float4*)(Bp + p);
        float gv[4] = { gray4(r4.x, g4.x, b4.x), gray4(r4.y, g4.y, b4.y),
                        gray4(r4.z, g4.z, b4.z), gray4(r4.w, g4.w, b4.w) };
#pragma unroll
        for (int j = 0; j < 4; ++j) {
            float im = (gv[j] - mean) * istd;
            float ab = fabsf(im);
            float fx = fx0 + (float)j;
            a0 += im;
            a1 += fx * im;
            a2 += fy * im;
            a3 += fx * fx * im;
            a4 += fx * fy * im;
            a5 += fy * fy * im;
            a6 += ab;
            a7 += fx * ab;
            a8 += fy * ab;
        }
    }
    float acc[9] = { a0, a1, a2, a3, a4, a5, a6, a7, a8 };
    float* o = partB + ((size_t)img * NBLK + blockIdx.x) * 9;
#pragma unroll
    for (int c = 0; c < 9; ++c) {
        float r = blockReduce(acc[c], smem);
        if (threadIdx.x == 0) o[c] = r;
    }
}

// ---------------------------------------------------------------------------
// Params: reduce 9 moments, closed-form 2x2 eigensolve, corner max_r.
// params[img*8] = { cx, cy, rc, rs, inv_max_r, inv_denom, mean, istd }
// ---------------------------------------------------------------------------
__global__ void k_params(const float* __restrict__ partB, const float* __restrict__ stats,
                         float* __restrict__ params) {
    __shared__ float smem[2];
    const int img = blockIdx.x;
    const float* p = partB + (size_t)img * NBLK * 9;
    float red[9];
#pragma unroll
    for (int c = 0; c < 9; ++c)
        red[c] = blockReduce(p[threadIdx.x * 9 + c], smem);
    if (threadIdx.x != 0) return;

    const float S   = red[0], Sx  = red[1], Sy  = red[2];
    const float Sxx = red[3], Sxy = red[4], Syy = red[5];
    const float Sa  = red[6], Sax = red[7], Say = red[8];

    const float safe = fmaxf(Sa, 1e-8f);
    const float cy = (Sa < 1e-8f) ? (HH * 0.5f) : (Say / safe);
    const float cx = (Sa < 1e-8f) ? (WW * 0.5f) : (Sax / safe);

    // inertia about centroid (of the signed normalized image)
    const float ixx = Sxx - 2.f * cx * Sx + cx * cx * S;
    const float ixy = Sxy - cx * Sy - cy * Sx + cx * cy * S;
    const float iyy = Syy - 2.f * cy * Sy + cy * cy * S;

    // smallest-eigenvalue eigenvector (sign ambiguity cancels in |moments|)
    const float tr = ixx + iyy, df = ixx - iyy;
    const float d  = sqrtf(df * df + 4.f * ixy * ixy);
    const float l0 = 0.5f * (tr - d);
    float vx = ixy, vy = l0 - ixx;
    if (vx == 0.f && vy == 0.f) { vx = l0 - iyy; vy = ixy; }
    if (vx == 0.f && vy == 0.f) { vx = 1.f; vy = 0.f; }
    const float invn = 1.f / sqrtf(vx * vx + vy * vy);
    const float rc = vx * invn;      // cos(angle)
    const float rs = vy * invn;      // sin(angle)

    // max r^2 is attained at an image corner (rotation preserves radius)
    const float dx = fmaxf(cx, (float)(WW - 1) - cx);
    const float dy = fmaxf(cy, (float)(HH - 1) - cy);
    const float max_r = sqrtf(dx * dx + dy * dy) + 1e-8f;

    float* o = params + img * 8;
    o[0] = cx; o[1] = cy; o[2] = rc; o[3] = rs;
    o[4] = 1.f / max_r;
    o[5] = 1.f / (max_r * max_r + 1e-8f);
    o[6] = stats[img * 2 + 0];
    o[7] = stats[img * 2 + 1];
}

// ---------------------------------------------------------------------------
// Pass C: Zernike accumulation. 10 unique accumulators (±m pairs share
// magnitudes; radial polys are R00=1, R11=r, R22=r^2, R20=2r^2-1,
// R33=r^3, R31=3r^3-2r). Partials padded to 16 channels for the WMMA reduce.
// ---------------------------------------------------------------------------
__global__ void k_zernike(const float* __restrict__ x, const float* __restrict__ params,
                          float* __restrict__ partC) {
    __shared__ float smem[TPB / 32];
    const int img = blockIdx.y;
    const float* q = params + img * 8;
    const float cx = q[0], cy = q[1], rc = q[2], rs = q[3];
    const float inv_max_r = q[4];
    const float mean = q[6], istd = q[7];

    const float* Rp = x + (size_t)img * 3 * NPX;
    const float* Gp = Rp + NPX;
    const float* Bp = Rp + 2 * NPX;
    const int base = blockIdx.x * PXB;

    float m0 = 0.f, m1 = 0.f, m2 = 0.f, m3 = 0.f, m4 = 0.f,
          m5 = 0.f, m6 = 0.f, m7 = 0.f, m8 = 0.f, m9 = 0.f;
#pragma unroll
    for (int i = 0; i < 4; ++i) {
        const int p = base + i * 1024 + threadIdx.x * 4;
        const float y  = (float)(p >> 9) - cy;
        const float x0 = (float)(p & 511) - cx;
        float4 r4 = *(const float4*)(Rp + p);
        float4 g4 = *(const float4*)(Gp + p);
        float4 b4 = *(const float4*)(Bp + p);
        float gv[4] = { gray4(r4.x, g4.x, b4.x), gray4(r4.y, g4.y, b4.y),
                        gray4(r4.z, g4.z, b4.z), gray4(r4.w, g4.w, b4.w) };
#pragma unroll
        for (int j = 0; j < 4; ++j) {
            const float im = (gv[j] - mean) * istd;
            const float xx = x0 + (float)j;
            // rotate by -angle: xr = x*cos a + y*sin a ; yr = y*cos a - x*sin a
            const float xr = xx * rc + y * rs;
            const float yr = y * rc - xx * rs;
            const float r2  = xr * xr + yr * yr;
            const float rho = sqrtf(r2 + 1e-20f);
            const float r   = fminf(rho * inv_max_r, 1.f);
            const float irho = 1.f / rho;
            const float c1 = xr * irho, s1 = yr * irho;
            const float c2 = c1 * c1 - s1 * s1, s2 = 2.f * c1 * s1;
            const float c3 = c2 * c1 - s2 * s1, s3 = s2 * c1 + c2 * s1;
            const float rr = r * r, rrr = rr * r;
            const float R20 = 2.f * rr - 1.f;
            const float R31 = 3.f * rrr - 2.f * r;
            m0 += im;
            const float t1 = im * r;   m1 += t1 * c1; m2 += t1 * s1;
            const float t2 = im * rr;  m3 += t2 * c2; m4 += t2 * s2;
            m5 += im * R20;
            const float t3 = im * rrr; m6 += t3 * c3; m7 += t3 * s3;
            const float t4 = im * R31; m8 += t4 * c1; m9 += t4 * s1;
        }
    }
    float acc[10] = { m0, m1, m2, m3, m4, m5, m6, m7, m8, m9 };
    float* o = partC + ((size_t)img * NBLK + blockIdx.x) * 16;
#pragma unroll
    for (int c = 0; c < 10; ++c) {
        float r = blockReduce(acc[c], smem);
        if (threadIdx.x == 0) o[c] = r;
    }
    if (threadIdx.x == 0) {
        o[10] = 0.f; o[11] = 0.f; o[12] = 0.f;
        o[13] = 0.f; o[14] = 0.f; o[15] = 0.f;
    }
}

// ---------------------------------------------------------------------------
// WMMA reduce + finalize. One wave per image.
// Exact fp32 tree sum of 64 block-partials x 16 channels using
// V_WMMA_F32_16X16X4_F32:  C += A(16x4 = partials^T chunk) * Ones(4x16).
// A layout (ISA 7.12.2): lanes 0-15 hold K=0,1 (VGPR0,1); lanes 16-31 hold K=2,3;
// M = lane % 16. B = all-ones is layout agnostic.
// D layout: VGPR j -> lanes 0-15: M=j, lanes 16-31: M=j+8 (all N columns equal).
// ---------------------------------------------------------------------------
__global__ void k_wmma_finalize(const float* __restrict__ partC,
                                const float* __restrict__ params,
                                float* __restrict__ out) {
    const int img  = blockIdx.x;
    const int lane = threadIdx.x;       // 32 threads = 1 wave, EXEC all ones
    const float* P = partC + (size_t)img * NBLK * 16;

    const int m  = lane & 15;
    const int kb = (lane >> 4) << 1;    // K-pair base: 0 for lanes<16, 2 otherwise

    v2f ones; ones[0] = 1.f; ones[1] = 1.f;
    v8f c = {};
#pragma unroll
    for (int t = 0; t < NBLK / 4; ++t) {
        const int b = t * 4 + kb;
        v2f a;
        a[0] = P[(size_t)b * 16 + m];
        a[1] = P[(size_t)(b + 1) * 16 + m];
        c = __builtin_amdgcn_wmma_f32_16x16x4_f32(
                /*neg_a=*/false, a, /*neg_b=*/false, ones,
                /*c_mod=*/(short)0, c, /*reuse_a=*/false, /*reuse_b=*/false);
    }
    // lane 0 holds channels 0..7 in c[0..7]; lane 16 holds channels 8..15.
    const float ch8 = __shfl(c[0], 16, 32);
    const float ch9 = __shfl(c[1], 16, 32);

    if (lane == 0) {
        const float inv_denom = params[img * 8 + 5];
        const float f0  = fabsf(c[0]) * inv_denom;                       // (0,0)
        const float f12 = sqrtf(c[1] * c[1] + c[2] * c[2]) * inv_denom;  // (1,+-1)
        const float f35 = sqrtf(c[3] * c[3] + c[4] * c[4]) * inv_denom;  // (2,+-2)
        const float f4  = fabsf(c[5]) * inv_denom;                       // (2,0)
        const float f69 = sqrtf(c[6] * c[6] + c[7] * c[7]) * inv_denom;  // (3,+-3)
        const float f78 = sqrtf(ch8 * ch8 + ch9 * ch9) * inv_denom;      // (3,+-1)
        float* o = out + img * 20;
        o[0] = f0;  o[1] = f12; o[2] = f12; o[3] = f35; o[4] = f4;
        o[5] = f35; o[6] = f69; o[7] = f78; o[8] = f78; o[9] = f69;
#pragma unroll
        for (int j = 10; j < 20; ++j) o[j] = 0.f;
    }
}

// ---------------------------------------------------------------------------
extern "C" void kernel_launch(void* const* d_in, const int* in_sizes, int n_in,
                              void* d_out, int out_size, void* d_ws, size_t ws_size,
                              hipStream_t stream) {
    (void)in_sizes; (void)n_in; (void)out_size; (void)ws_size;
    const float* x = (const float*)d_in[0];
    float* out = (float*)d_out;
    float* ws  = (float*)d_ws;

    // workspace layout (floats)
    float* partA  = ws;                              // 32*64*2   = 4096
    float* stats  = partA  + BIMG * NBLK * 2;        // 32*2      = 64
    float* partB  = stats  + BIMG * 2;               // 32*64*9   = 18432
    float* params = partB  + BIMG * NBLK * 9;        // 32*8      = 256
    float* partC  = params + BIMG * 8;               // 32*64*16  = 32768

    dim3 pgrid(NBLK, BIMG);
    k_gray_stats<<<pgrid, TPB, 0, stream>>>(x, partA);
    k_stats<<<BIMG, 64, 0, stream>>>(partA, stats);
    k_moments<<<pgrid, TPB, 0, stream>>>(x, stats, partB);
    k_params<<<BIMG, 64, 0, stream>>>(partB, stats, params);
    k_zernike<<<pgrid, TPB, 0, stream>>>(x, params, partC);
    k_wmma_finalize<<<BIMG, 32, 0, stream>>>(partC, params, out);
}